// AUM_30743375905337
// MI455X (gfx1250) — compile-verified
//
#include <hip/hip_runtime.h>
#include <hip/hip_bf16.h>

typedef __attribute__((ext_vector_type(2))) float v2f;
typedef __attribute__((ext_vector_type(8))) float v8f;
typedef __attribute__((ext_vector_type(4))) unsigned int u32x4;
typedef __attribute__((ext_vector_type(8))) int i32x8;
typedef __attribute__((ext_vector_type(4))) int i32x4;

__device__ __forceinline__ int imin(int a, int b) { return a < b ? a : b; }
__device__ __forceinline__ int imax(int a, int b) { return a > b ? a : b; }

// gfx1250 async memory->LDS copy (ASYNCcnt path).
__device__ __forceinline__ void async_b32_to_lds(unsigned lds_addr, const float* g) {
    asm volatile("global_load_async_to_lds_b32 %0, %1, off"
                 :: "v"(lds_addr), "v"((unsigned long long)(uintptr_t)g)
                 : "memory");
}
__device__ __forceinline__ unsigned lds_off(const volatile void* p) {
    return (unsigned)(uintptr_t)p;   // addrspace(3) offset survives in low 32 bits
}

// ---------------------------------------------------------------------------
// K1: BN stats via Tensor Data Mover.  One block per channel.  Wave 0 issues a
// single TDM DMA of the channel's [4 x 4096] 2D tile (dim0 contiguous 4096,
// dim1 stride 256*4096) into LDS, waits TENSORcnt, then the block reduces.
// ---------------------------------------------------------------------------
__global__ void k_bnstats(const float* __restrict__ high,
                          const float* __restrict__ gamma,
                          const float* __restrict__ beta,
                          float* __restrict__ scale,
                          float* __restrict__ shift) {
    __shared__ float buf[4 * 4096];      // 64 KB of the WGP's 320 KB LDS
    __shared__ float ssum[256];
    __shared__ float ssq[256];
    const int c = blockIdx.x;
    const int t = threadIdx.x;

    if (t < 32) {  // one wave issues the DMA (wave-uniform branch)
        const unsigned long long ga =
            (unsigned long long)(uintptr_t)(high + (size_t)c * 4096);
        u32x4 g0;
        g0[0] = 1u;                                   // count=1 user descriptor
        g0[1] = lds_off(&buf[0]);                     // lds_addr
        g0[2] = (unsigned)ga;                         // global_addr[31:0]
        g0[3] = ((unsigned)(ga >> 32) & 0x01FFFFFFu)  // global_addr[56:32]
                | 0x80000000u;                        // type=2 (image)
        i32x8 g1;
        g1[0] = (int)(2u << 16);                      // data_size=4B
        g1[1] = (int)(4096u << 16);                   // tensor_dim0[15:0]=4096
        g1[2] = (int)(4u << 16);                      // tensor_dim0 hi | tensor_dim1=4
        g1[3] = (int)(4096u << 16);                   // tensor_dim1 hi | tile_dim0=4096
        g1[4] = 4;                                    // tile_dim1=4, tile_dim2=0
        g1[5] = (int)(256u * 4096u);                  // tensor_dim0_stride
        g1[6] = 0;
        g1[7] = 0;
        i32x4 z4 = {0, 0, 0, 0};
        i32x8 z8 = {0, 0, 0, 0, 0, 0, 0, 0};
        __builtin_amdgcn_tensor_load_to_lds(g0, g1, z4, z4, z8, 0);
        __builtin_amdgcn_s_wait_tensorcnt(0);
    }
    __syncthreads();

    float s = 0.f, q = 0.f;
    for (int j = t; j < 4 * 4096; j += 256) {
        const float v = buf[j];
        s += v;
        q += v * v;
    }
    ssum[t] = s;
    ssq[t]  = q;
    __syncthreads();
    for (int st = 128; st > 0; st >>= 1) {
        if (t < st) { ssum[t] += ssum[t + st]; ssq[t] += ssq[t + st]; }
        __syncthreads();
    }
    if (t == 0) {
        const float mean = ssum[0] * (1.f / 16384.f);
        const float var  = ssq[0] * (1.f / 16384.f) - mean * mean;
        const float rs   = rsqrtf(var + 1e-5f);
        const float sc   = gamma[c] * rs;
        scale[c] = sc;
        shift[c] = beta[c] - mean * sc;
    }
}

// ---------------------------------------------------------------------------
// K2: down[b,c,ly,lx] = 0.25*sum(high 2x2)*scale[c] + shift[c]
// ---------------------------------------------------------------------------
__global__ void k_down(const float* __restrict__ high,
                       const float* __restrict__ scale,
                       const float* __restrict__ shift,
                       float* __restrict__ down) {
    const int idx = blockIdx.x * 256 + threadIdx.x;
    const int b   = idx >> 18;
    const int c   = (idx >> 10) & 255;
    const int ly  = (idx >> 5) & 31;
    const int lx  = idx & 31;
    const float* hp = high + (((size_t)(b * 256 + c)) * 64 + 2 * ly) * 64 + 2 * lx;
    const float s = hp[0] + hp[1] + hp[64] + hp[65];
    down[idx] = 0.25f * s * scale[c] + shift[c];
}

// ---------------------------------------------------------------------------
// K3: scores + softmax -> att[b,9,64,64].  16x16 pixel tile per block.
// Double-buffered async LDS staging: channel c+1's halos are in flight while
// channel c is accumulated; s_wait_asynccnt 3 retires the oldest 3 copies
// (in-order completion).  Border zero-padding applied via channel-invariant
// per-thread {0,1} tap masks, so all async loads use clamped valid addresses.
// ---------------------------------------------------------------------------
__global__ void k_scores(const float* __restrict__ high,
                         const float* __restrict__ down,
                         const float* __restrict__ scale,
                         const float* __restrict__ shift,
                         const float* __restrict__ bias,
                         float* __restrict__ att) {
    __shared__ float hs[2][512];   // 18*18 = 324 used per buffer (raw high)
    __shared__ float dsl[2][256];  // 10*10 = 100 used per buffer
    __shared__ float sc[256];
    __shared__ float sh[256];

    const int t  = threadIdx.x;
    const int b  = blockIdx.x >> 4;
    const int ty = (blockIdx.x >> 2) & 3;
    const int tx = blockIdx.x & 3;
    const int y0 = ty * 16, x0 = tx * 16;
    const int cy0 = ty * 8, cx0 = tx * 8;
    const int py = t >> 4, px = t & 15;
    const int y = y0 + py, x = x0 + px;
    const int cy = y >> 1, cx = x >> 1;

    sc[t] = scale[t];
    sh[t] = shift[t];

    // channel-invariant zero-padding masks, one per tap
    float mh[9], md[9];
#pragma unroll
    for (int di = -1; di <= 1; ++di)
#pragma unroll
        for (int dj = -1; dj <= 1; ++dj) {
            const int k = (di + 1) * 3 + dj + 1;
            mh[k] = (y + di >= 0 && y + di < 64 && x + dj >= 0 && x + dj < 64) ? 1.f : 0.f;
            md[k] = (cy + di >= 0 && cy + di < 32 && cx + dj >= 0 && cx + dj < 32) ? 1.f : 0.f;
        }

    // clamped source coordinates for the async halo loads
    int hsrc[2];
#pragma unroll
    for (int r = 0; r < 2; ++r) {
        const int idx = t + r * 256;                    // 0..511 (>=324 = padding)
        const int row = idx / 18, col = idx - row * 18;
        const int gy = imin(imax(y0 - 1 + row, 0), 63);
        const int gx = imin(imax(x0 - 1 + col, 0), 63);
        hsrc[r] = gy * 64 + gx;
    }
    int dsrc;
    {
        const int row = t / 10, col = t - row * 10;     // t>=100 = padding
        const int gy = imin(imax(cy0 - 1 + row, 0), 31);
        const int gx = imin(imax(cx0 - 1 + col, 0), 31);
        dsrc = gy * 32 + gx;
    }
    unsigned hd0[2], hd1[2], dd[2];
#pragma unroll
    for (int r = 0; r < 2; ++r) {
        hd0[r] = lds_off(&hs[r][t]);
        hd1[r] = lds_off(&hs[r][t + 256]);
        dd[r]  = lds_off(&dsl[r][t]);
    }

    __syncthreads();

    float acc[9];
#pragma unroll
    for (int k = 0; k < 9; ++k) acc[k] = 0.f;

    // prologue: channel 0 into buffer 0
    {
        const float* hc = high + ((size_t)(b * 256)) * 4096;
        const float* dc = down + ((size_t)(b * 256)) * 1024;
        async_b32_to_lds(hd0[0], hc + hsrc[0]);
        async_b32_to_lds(hd1[0], hc + hsrc[1]);
        async_b32_to_lds(dd[0],  dc + dsrc);
    }

    for (int c = 0; c < 256; ++c) {
        const int cb = c & 1, nb = cb ^ 1;
        if (c + 1 < 256) {
            const float* hc = high + ((size_t)(b * 256 + c + 1)) * 4096;
            const float* dc = down + ((size_t)(b * 256 + c + 1)) * 1024;
            __builtin_prefetch(hc + 4096 + y * 64 + x, 0, 0);  // warm L2 for c+2
            async_b32_to_lds(hd0[nb], hc + hsrc[0]);
            async_b32_to_lds(hd1[nb], hc + hsrc[1]);
            async_b32_to_lds(dd[nb],  dc + dsrc);
            asm volatile("s_wait_asynccnt 3" ::: "memory");  // channel c's 3 done
        } else {
            asm volatile("s_wait_asynccnt 0" ::: "memory");
        }
        __syncthreads();

        const float scv = sc[c], shv = sh[c];
        const int hy = py + 1, hxl = px + 1;
        const int dy = (py >> 1) + 1, dxl = (px >> 1) + 1;
#pragma unroll
        for (int di = -1; di <= 1; ++di)
#pragma unroll
            for (int dj = -1; dj <= 1; ++dj) {
                const int k = (di + 1) * 3 + dj + 1;
                const float hv = (hs[cb][(hy + di) * 18 + hxl + dj] * scv + shv) * mh[k];
                const float dv = dsl[cb][(dy + di) * 10 + dxl + dj] * md[k];
                const float df = hv - dv;
                acc[k] += df * df;
            }
        __syncthreads();
    }

    // bias + softmax over K=9
    float m = -3.402823466e+38f;
#pragma unroll
    for (int k = 0; k < 9; ++k) { acc[k] += bias[k]; m = fmaxf(m, acc[k]); }
    float s = 0.f;
#pragma unroll
    for (int k = 0; k < 9; ++k) { acc[k] = __expf(acc[k] - m); s += acc[k]; }
    const float inv = 1.f / s;
#pragma unroll
    for (int k = 0; k < 9; ++k)
        att[(((size_t)(b * 9 + k)) * 64 + y) * 64 + x] = acc[k] * inv;
}

// ---------------------------------------------------------------------------
// K4: aggregation as dense 16x16x16 WMMA.  The 3x3 windows of a 2x2 low-res
// block union to a 4x4 = 16-tap grid, so per block of 4 low-res pixels:
//   out[16ch x 16hp] = A[16ch x 16taps] * B[16taps x 16hp]
// B[tap,hp] = att weight if tap is inside hp's 3x3 window else 0 (built once
// per wave, channel-invariant).  4 chunks of V_WMMA_F32_16X16X4_F32 per
// 16-channel tile; all 16 N columns valid -> unconditional clause-batched
// stores.  Channels split 4-ways across waves for occupancy.
// ---------------------------------------------------------------------------
__global__ void k_agg(const float* __restrict__ low,
                      const float* __restrict__ att,
                      float* __restrict__ out) {
    const int lane = threadIdx.x & 31;
    const int wid  = blockIdx.x * (blockDim.x >> 5) + (threadIdx.x >> 5); // 0..4095
    const int b    = wid >> 10;
    const int blk  = (wid >> 2) & 255;  // 2x2 low-res block id (16x16 grid)
    const int mq   = wid & 3;           // channel quarter: 8 M-tiles of 16
    const int ly0  = (blk >> 4) * 2;
    const int lx0  = (blk & 15) * 2;

    const int half = lane >> 4;         // K-row pair selector within each chunk
    const int lid  = lane & 15;

    // N column -> high-res pixel: (oy,ox) low-res offset in block, (sy,sx) child
    const int oy = (lid >> 3) & 1, ox = (lid >> 2) & 1;
    const int sy = (lid >> 1) & 1, sx = lid & 1;
    const int hy = (ly0 + oy) * 2 + sy;
    const int hx = (lx0 + ox) * 2 + sx;

    const float* attb = att + (size_t)b * 9 * 4096;
    const float* lowb = low + (size_t)b * 512 * 1024;
    float*       outb = out + (size_t)b * 512 * 4096;
    const unsigned hpoff = (unsigned)(hy * 64 + hx);

    // B fragments over the 16-tap union grid (channel-invariant)
    v2f bfrag[4];
#pragma unroll
    for (int kc = 0; kc < 4; ++kc) {
#pragma unroll
        for (int u = 0; u < 2; ++u) {
            const int T  = kc * 4 + half * 2 + u;
            const int di = (T >> 2) - oy - 1;
            const int dj = (T & 3) - ox - 1;
            const bool ok = ((unsigned)(di + 1) <= 2u) & ((unsigned)(dj + 1) <= 2u);
            const int k  = ok ? (di + 1) * 3 + (dj + 1) : 0;
            const float v = attb[(unsigned)(k * 4096) + hpoff];
            ((float*)&bfrag[kc])[u] = ok ? v : 0.f;
        }
    }

    // per-lane A tap offsets / validity (mt-invariant); clamped addresses
    unsigned toff[8];
    bool     tok[8];
#pragma unroll
    for (int kc = 0; kc < 4; ++kc)
#pragma unroll
        for (int u = 0; u < 2; ++u) {
            const int T  = kc * 4 + half * 2 + u;
            const int yy = ly0 - 1 + (T >> 2);
            const int xx = lx0 - 1 + (T & 3);
            tok[kc * 2 + u]  = (yy >= 0) & (yy < 32) & (xx >= 0) & (xx < 32);
            toff[kc * 2 + u] = (unsigned)(imin(imax(yy, 0), 31) * 32 +
                                          imin(imax(xx, 0), 31));
        }

    for (int mt = mq * 8; mt < mq * 8 + 8; ++mt) {
        const unsigned chbase = (unsigned)((mt * 16 + lid) * 1024);
        v8f acc = {};
#pragma unroll
        for (int kc = 0; kc < 4; ++kc) {
            const float a0 = lowb[chbase + toff[kc * 2 + 0]];
            const float a1 = lowb[chbase + toff[kc * 2 + 1]];
            v2f a;
            a.x = tok[kc * 2 + 0] ? a0 : 0.f;
            a.y = tok[kc * 2 + 1] ? a1 : 0.f;
            // (neg_a, A, neg_b, B, c_mod, C, reuse_a, reuse_b)
            acc = __builtin_amdgcn_wmma_f32_16x16x4_f32(
                false, a, false, bfrag[kc], (short)0, acc, false, false);
        }
        // D: VGPR v holds M=v (lanes<16) / M=8+v (lanes>=16), N=lid
        const unsigned obase = (unsigned)((mt * 16 + half * 8) * 4096) + hpoff;
#pragma unroll
        for (int v = 0; v < 8; ++v) outb[obase + (unsigned)v * 4096] = acc[v];
    }
}

// ---------------------------------------------------------------------------
extern "C" void kernel_launch(void* const* d_in, const int* in_sizes, int n_in,
                              void* d_out, int out_size, void* d_ws, size_t ws_size,
                              hipStream_t stream) {
    const float* low   = (const float*)d_in[0]; // [4,512,32,32]
    const float* high  = (const float*)d_in[1]; // [4,256,64,64]
    const float* gamma = (const float*)d_in[2]; // [256]
    const float* beta  = (const float*)d_in[3]; // [256]
    const float* bias  = (const float*)d_in[4]; // [9]
    float* out = (float*)d_out;                 // [4,512,64,64]

    float* ws    = (float*)d_ws;
    float* scale = ws;                          // 256
    float* shift = ws + 256;                    // 256
    float* down  = ws + 512;                    // 4*256*32*32 = 1,048,576
    float* att   = down + 4 * 256 * 32 * 32;    // 4*9*64*64   =   147,456

    k_bnstats<<<256, 256, 0, stream>>>(high, gamma, beta, scale, shift);
    k_down<<<4096, 256, 0, stream>>>(high, scale, shift, down);
    k_scores<<<64, 256, 0, stream>>>(high, down, scale, shift, bias, att);
    k_agg<<<512, 256, 0, stream>>>(low, att, out);
}